// CosAttention_9216999817912
// MI455X (gfx1250) — compile-verified
//
#include <hip/hip_runtime.h>
#include <math.h>

#define B 4
#define T 8192
#define D 64
#define P 8
#define CHUNK 128
#define NC (T / CHUNK) /* 64 */
#define PI_F 3.14159265358979323846f
#define EPS_F 1e-3f

typedef __attribute__((ext_vector_type(2))) float v2f;
typedef __attribute__((ext_vector_type(8))) float v8f;

// ---------------------------------------------------------------------------
// Kernel 1: q/k projections (b,t,p), tanh squash, trig -> cq/sq/ck/sk scratch
// ---------------------------------------------------------------------------
__global__ __launch_bounds__(256) void k_proj(
    const float* __restrict__ query, const float* __restrict__ WQ,
    const float* __restrict__ WK,
    float* __restrict__ cq, float* __restrict__ sq,
    float* __restrict__ ck, float* __restrict__ sk)
{
    __shared__ float sWQ[P * D];
    __shared__ float sWK[P * D];
    int tid = threadIdx.x;
    for (int i = tid; i < P * D; i += 256) { sWQ[i] = WQ[i]; sWK[i] = WK[i]; }
    __syncthreads();

    int row = blockIdx.x * 256 + tid;             // 0 .. B*T-1
    const float* qr = query + (size_t)row * D;

    float aq[P], ak[P];
#pragma unroll
    for (int p = 0; p < P; ++p) { aq[p] = 0.f; ak[p] = 0.f; }
    for (int i = 0; i < D; ++i) {
        float x = qr[i];
#pragma unroll
        for (int p = 0; p < P; ++p) {
            aq[p] = fmaf(x, sWQ[p * D + i], aq[p]);
            ak[p] = fmaf(x, sWK[p * D + i], ak[p]);
        }
    }
#pragma unroll
    for (int p = 0; p < P; ++p) {
        float qv = tanhf(aq[p]) * (PI_F * 0.25f);
        float kv = tanhf(ak[p]) * (PI_F * 0.25f);
        size_t o = (size_t)row * P + p;
        cq[o] = cosf(qv); sq[o] = sinf(qv);
        ck[o] = cosf(kv); sk[o] = sinf(kv);
    }
}

// ---------------------------------------------------------------------------
// Kernel 2: per-(b,chunk) sums of ck*query, sk*query (64-vec per p) + scalar
// sums of ck, sk. 64 threads: thread i owns dim i.
// ---------------------------------------------------------------------------
__global__ __launch_bounds__(64) void k_chunk_sums(
    const float* __restrict__ query,
    const float* __restrict__ ck, const float* __restrict__ sk,
    float* __restrict__ vecC, float* __restrict__ vecS,
    float* __restrict__ sclC, float* __restrict__ sclS)
{
    int blk = blockIdx.x;              // b*NC + c
    int b = blk / NC, c = blk % NC;
    int i = threadIdx.x;               // 0..63

    float aC[P], aS[P], kc[P], ks[P];
#pragma unroll
    for (int p = 0; p < P; ++p) { aC[p] = 0.f; aS[p] = 0.f; kc[p] = 0.f; ks[p] = 0.f; }

    int t0 = c * CHUNK;
    for (int t = 0; t < CHUNK; ++t) {
        size_t row = (size_t)b * T + t0 + t;
        float x = query[row * D + i];
#pragma unroll
        for (int p = 0; p < P; ++p) {
            float cv = ck[row * P + p];
            float sv = sk[row * P + p];
            aC[p] = fmaf(cv, x, aC[p]);
            aS[p] = fmaf(sv, x, aS[p]);
            kc[p] += cv; ks[p] += sv;
        }
    }
#pragma unroll
    for (int p = 0; p < P; ++p) {
        size_t vi = (((size_t)(b * P + p)) * NC + c) * D + i;
        vecC[vi] = aC[p]; vecS[vi] = aS[p];
        if (i == 0) {
            size_t si = (size_t)(b * P + p) * NC + c;
            sclC[si] = kc[p]; sclS[si] = ks[p];
        }
    }
}

// ---------------------------------------------------------------------------
// Kernel 3: sequential exclusive scan over chunks per (b,p), in place.
// ---------------------------------------------------------------------------
__global__ __launch_bounds__(160) void k_scan(
    float* __restrict__ vecC, float* __restrict__ vecS,
    float* __restrict__ sclC, float* __restrict__ sclS)
{
    int bp = blockIdx.x;               // b*P + p
    int tid = threadIdx.x;
    if (tid < 64) {
        float run = 0.f;
        for (int c = 0; c < NC; ++c) {
            size_t idx = (((size_t)bp) * NC + c) * D + tid;
            float v = vecC[idx]; vecC[idx] = run; run += v;
        }
    } else if (tid < 128) {
        int i = tid - 64;
        float run = 0.f;
        for (int c = 0; c < NC; ++c) {
            size_t idx = (((size_t)bp) * NC + c) * D + i;
            float v = vecS[idx]; vecS[idx] = run; run += v;
        }
    } else if (tid == 128) {
        float run = 0.f;
        for (int c = 0; c < NC; ++c) {
            size_t idx = (size_t)bp * NC + c;
            float v = sclC[idx]; sclC[idx] = run; run += v;
        }
    } else if (tid == 129) {
        float run = 0.f;
        for (int c = 0; c < NC; ++c) {
            size_t idx = (size_t)bp * NC + c;
            float v = sclS[idx]; sclS[idx] = run; run += v;
        }
    }
}

// ---------------------------------------------------------------------------
// Kernel 4: block per (b,chunk), 4 waves. Head-outer / subtile-inner:
//  - W_V fragments register-resident per head (loaded once, reused 8x)
//  - scan carries register-resident in threads 0..63
//  - query+trig staged once per block (whole 128-row chunk)
//  - 8 live WMMA accumulator tiles summed over all heads
// ---------------------------------------------------------------------------
__global__ __launch_bounds__(128) void k_main(
    const float* __restrict__ query,
    const float* __restrict__ WV,      // (P, D, D)
    const float* __restrict__ WO,      // (P)
    const float* __restrict__ cq, const float* __restrict__ sq,
    const float* __restrict__ ck, const float* __restrict__ sk,
    const float* __restrict__ vecC, const float* __restrict__ vecS,
    const float* __restrict__ sclC, const float* __restrict__ sclS,
    float* __restrict__ out)
{
    __shared__ float sQ[CHUNK * D];    // 32 KB: whole chunk of query rows
    __shared__ float sCQ[CHUNK * P];   // 4 KB each: per-row trig values
    __shared__ float sSQ[CHUNK * P];
    __shared__ float sCK[CHUNK * P];
    __shared__ float sSK[CHUNK * P];
    __shared__ float sY[2][16 * D];    // 8 KB: double-buffered Y tile

    int blk = blockIdx.x;              // b*NC + c
    int b = blk / NC, c = blk % NC;
    int tid = threadIdx.x;
    int lane = tid & 31;
    int wave = tid >> 5;
    int lhalf = lane >> 4;
    int lmod = lane & 15;
    int nb = wave * 16;                // this wave's output-column tile

    size_t rowbase = (size_t)b * T + (size_t)c * CHUNK;

    for (int idx = tid; idx < CHUNK * D; idx += 128)
        sQ[idx] = query[(rowbase + idx / D) * D + (idx % D)];
    for (int idx = tid; idx < CHUNK * P; idx += 128) {
        size_t o = (rowbase + idx / P) * P + (idx % P);
        sCQ[idx] = cq[o]; sSQ[idx] = sq[o];
        sCK[idx] = ck[o]; sSK[idx] = sk[o];
    }
    __syncthreads();

    v8f acc[CHUNK / 16];               // 8 x (16x16) fp32 accumulator tiles
#pragma unroll
    for (int st = 0; st < CHUNK / 16; ++st) acc[st] = (v8f){};

    for (int p = 0; p < P; ++p) {
        float wo = WO[p];

        // B fragments (ISA layout: lane -> N=l%16, VGPR j -> K=j+2*(l/16))
        const float* wvp = WV + (size_t)p * D * D;
        v2f bf[16];
#pragma unroll
        for (int kk = 0; kk < 16; ++kk) {
            int k0 = 4 * kk + 2 * lhalf;
            bf[kk][0] = wvp[(size_t)k0 * D + nb + lmod];
            bf[kk][1] = wvp[(size_t)(k0 + 1) * D + nb + lmod];
        }

        // scan carries for this head, register-resident in threads 0..63
        float cc = 0.f, ss = 0.f, kc = 0.f, ks = 0.f;
        if (tid < D) {
            size_t vi = (((size_t)(b * P + p)) * NC + c) * D + tid;
            cc = vecC[vi]; ss = vecS[vi];
            size_t si = (size_t)(b * P + p) * NC + c;
            kc = sclC[si]; ks = sclS[si];
        }

#pragma unroll
        for (int st = 0; st < CHUNK / 16; ++st) {
            float* y = sY[st & 1];
            if (tid < D) {
                int i = tid;
#pragma unroll
                for (int r = 0; r < 16; ++r) {
                    int rg = st * 16 + r;
                    float cv = sCK[rg * P + p], sv = sSK[rg * P + p];
                    float x = sQ[rg * D + i];
                    cc = fmaf(cv, x, cc);      // inclusive cumsum(ck*query)
                    ss = fmaf(sv, x, ss);
                    kc += cv; ks += sv;        // inclusive cumsum(ck),(sk)
                    float cqv = sCQ[rg * P + p], sqv = sSQ[rg * P + p];
                    float denom = fmaf(cqv, kc, fmaf(sqv, ks, EPS_F));
                    float z = cqv * cc + sqv * ss;
                    y[r * D + i] = (wo / denom) * z;
                }
            }
            __syncthreads();
            // acc[st] += Y(16x64) x WV[p] slice, K split into 16 chunks of 4
#pragma unroll
            for (int kk = 0; kk < 16; ++kk) {
                int k0 = 4 * kk + 2 * lhalf;
                v2f a;
                a[0] = y[lmod * D + k0];
                a[1] = y[lmod * D + k0 + 1];
                acc[st] = __builtin_amdgcn_wmma_f32_16x16x4_f32(
                    false, a, false, bf[kk], (short)0, acc[st], false, false);
            }
        }
        __syncthreads();   // buffer handoff before next head rebuilds Y
    }

    // store: C layout VGPR j -> M = j + 8*(lane/16), N = nb + lane%16
#pragma unroll
    for (int st = 0; st < CHUNK / 16; ++st) {
#pragma unroll
        for (int j = 0; j < 8; ++j) {
            int row = st * 16 + j + 8 * lhalf;
            out[(rowbase + row) * D + nb + lmod] = acc[st][j];
        }
    }
}

// ---------------------------------------------------------------------------
extern "C" void kernel_launch(void* const* d_in, const int* in_sizes, int n_in,
                              void* d_out, int out_size, void* d_ws, size_t ws_size,
                              hipStream_t stream)
{
    const float* query = (const float*)d_in[0];
    const float* WQ = (const float*)d_in[1];
    const float* WK = (const float*)d_in[2];
    const float* WV = (const float*)d_in[3];
    const float* WO = (const float*)d_in[4];
    float* out = (float*)d_out;
    float* ws = (float*)d_ws;

    // workspace layout (floats): 4x trig (B*T*P each) + chunk prefixes
    float* cq = ws;
    float* sq = cq + (size_t)B * T * P;
    float* ck = sq + (size_t)B * T * P;
    float* sk = ck + (size_t)B * T * P;
    float* vecC = sk + (size_t)B * T * P;
    float* vecS = vecC + (size_t)B * P * NC * D;
    float* sclC = vecS + (size_t)B * P * NC * D;
    float* sclS = sclC + (size_t)B * P * NC;

    k_proj<<<(B * T) / 256, 256, 0, stream>>>(query, WQ, WK, cq, sq, ck, sk);
    k_chunk_sums<<<B * NC, 64, 0, stream>>>(query, ck, sk, vecC, vecS, sclC, sclS);
    k_scan<<<B * P, 160, 0, stream>>>(vecC, vecS, sclC, sclS);
    k_main<<<B * NC, 128, 0, stream>>>(query, WV, WO, cq, sq, ck, sk,
                                       vecC, vecS, sclC, sclS, out);
}